// MaskedConv2d_53867479826875
// MI455X (gfx1250) — compile-verified
//
#include <hip/hip_runtime.h>
#include <math.h>

typedef float v2f __attribute__((ext_vector_type(2)));
typedef float v8f __attribute__((ext_vector_type(8)));

#define B_    4
#define CIN   128
#define COUT  128
#define HW    256      // 16*16 pixels
#define AD    16       // attn dim
#define KDIM  1152     // CIN*9
#define WN    147456   // COUT*KDIM

#define BPAD  1156     // per-pixel LDS stride (1156 mod 64 == 4 -> conflict-free halves)
#define LDSB  (16 * BPAD * 4)   // dynamic LDS bytes for the im2col tile

// float-indexed offsets into d_ws
#define WS_S     0        // [B*HW]   scalar scores s_bp
#define WS_SHIFT 1024     // [B*HW]   softmax shifts
#define WS_INVD  2048     // [B*HW]   1/(256 * denom)
#define WS_MM    3072     // [2]      wmax, wmin
#define WS_MW    3584     // [B*COUT*KDIM] masked weights

// -------- Kernel 1: s_bp = ( x[b,:,p] . g + bias0 ) / 4 ----------------------
__global__ void k_sproj(const float* __restrict__ x,
                        const float* __restrict__ fqw,   // [AD, CIN]
                        const float* __restrict__ fqb,   // [AD]
                        const float* __restrict__ fkw,   // [AD] (fc_k_w[:,0])
                        float* __restrict__ ws) {
  __shared__ float g[CIN];
  const int b = blockIdx.x, t = threadIdx.x;
  if (t < CIN) {
    float acc = 0.f;
    for (int a = 0; a < AD; ++a) acc += fkw[a] * fqw[a * CIN + t];
    g[t] = acc;
  }
  __syncthreads();
  float bias0 = 0.f;
  for (int a = 0; a < AD; ++a) bias0 += fkw[a] * fqb[a];
  const float* xb = x + b * CIN * HW + t;   // t = pixel index
  float acc = bias0;
  for (int c = 0; c < CIN; ++c) acc += xb[c * HW] * g[c];
  ws[WS_S + b * HW + t] = 0.25f * acc;
}

// -------- Kernel 2: min/max of flat weight (softmax stabilization) -----------
__global__ void k_wminmax(const float* __restrict__ w, float* __restrict__ ws) {
  __shared__ float smx[256], smn[256];
  const int t = threadIdx.x;
  float mx = -3.402823466e38f, mn = 3.402823466e38f;
  for (int i = t; i < WN; i += 256) { float v = w[i]; mx = fmaxf(mx, v); mn = fminf(mn, v); }
  smx[t] = mx; smn[t] = mn; __syncthreads();
  for (int s = 128; s > 0; s >>= 1) {
    if (t < s) { smx[t] = fmaxf(smx[t], smx[t + s]); smn[t] = fminf(smn[t], smn[t + s]); }
    __syncthreads();
  }
  if (t == 0) { ws[WS_MM] = smx[0]; ws[WS_MM + 1] = smn[0]; }
}

// -------- Kernel 3: per (b,p) softmax denominator -> invD' = 1/(256*D) -------
__global__ void k_denom(const float* __restrict__ w, float* __restrict__ ws) {
  __shared__ float sm[256];
  const int bp = blockIdx.x, t = threadIdx.x;
  const float s = ws[WS_S + bp];
  const float wmax = ws[WS_MM], wmin = ws[WS_MM + 1];
  const float shift = (s >= 0.f) ? s * wmax : s * wmin;
  float acc = 0.f;
  for (int n = t; n < WN; n += 256) acc += __expf(w[n] * s - shift);
  sm[t] = acc; __syncthreads();
  for (int r = 128; r > 0; r >>= 1) { if (t < r) sm[t] += sm[t + r]; __syncthreads(); }
  if (t == 0) {
    ws[WS_SHIFT + bp] = shift;
    ws[WS_INVD + bp]  = 1.f / (256.f * sm[0]);
  }
}

// -------- Kernel 4: masked_w[b,n] = w_n + gamma*(m*(w_n*cvo+cvb)+fo_b) -------
__global__ void k_maskw(const float* __restrict__ w,
                        const float* __restrict__ fvw, const float* __restrict__ fvb,
                        const float* __restrict__ fow, const float* __restrict__ fob,
                        const float* __restrict__ gam,
                        float* __restrict__ ws) {
  __shared__ float ls[HW], lsh[HW], lid[HW];
  const int b = blockIdx.y, t = threadIdx.x;
  const int n = blockIdx.x * 256 + t;     // 576*256 == WN exactly
  ls[t]  = ws[WS_S     + b * HW + t];
  lsh[t] = ws[WS_SHIFT + b * HW + t];
  lid[t] = ws[WS_INVD  + b * HW + t];
  __syncthreads();
  const float wn = w[n];
  float acc = 0.f;
#pragma unroll 4
  for (int p = 0; p < HW; ++p) acc += __expf(wn * ls[p] - lsh[p]) * lid[p];  // m[b,n]
  float cvo = 0.f, cvb = 0.f;
  for (int a = 0; a < AD; ++a) { cvo += fvw[a] * fow[a]; cvb += fvb[a] * fow[a]; }
  const float mw = wn + gam[0] * (acc * (wn * cvo + cvb) + fob[0]);
  ws[WS_MW + b * WN + n] = mw;
}

// -------- Kernel 5: per-batch conv as implicit GEMM on V_WMMA_F32_16X16X4_F32
// Block = (pixel-row nt, batch b). A 16-pixel tile is exactly image row y=nt,
// so all 8 waves (one per cout tile) share one 1152x16 im2col tile, staged
// once into LDS (pixel-major, stride BPAD). Hot loop per wave:
//   global_load_b64 (A) + ds_load_b64 (B) + v_wmma_f32_16x16x4_f32.
__global__ void __launch_bounds__(256)
k_conv_wmma(const float* __restrict__ x,
            const float* __restrict__ ws,
            float* __restrict__ out) {
  extern __shared__ float sB[];          // [16][BPAD] im2col tile
  const int nt = blockIdx.x;             // image row / pixel tile (16)
  const int b  = blockIdx.y;             // batch (4)
  const int t  = threadIdx.x;
  const float* xb = x + b * CIN * HW;

  // ---- Stage im2col tile: sB[p][kc] = x[b][ci][nt+dy-1][p+dx-1] (0 if OOB)
  for (int p = 0; p < 16; ++p) {
    for (int kc = t; kc < KDIM; kc += 256) {
      const int ci = kc / 9;
      const int r  = kc - ci * 9;
      const int dy = r / 3;
      const int dx = r - dy * 3;
      const int yy = nt + dy - 1;
      const int xx = p + dx - 1;
      const bool in = ((unsigned)yy < 16u) && ((unsigned)xx < 16u);
      sB[p * BPAD + kc] = in ? xb[ci * HW + yy * 16 + xx] : 0.f;
    }
  }
  __syncthreads();

  // ---- WMMA loop: wave w owns cout tile mt = w
  const int mt   = t >> 5;               // 0..7
  const int lane = t & 31;
  const int half = lane >> 4;            // K-half select (ISA f32 A/B layout)
  const int l    = lane & 15;            // M (A rows) / N (B pixels)

  const float* aRow = ws + WS_MW + b * WN + (mt * 16 + l) * KDIM + 2 * half;
  const float* bRow = sB + l * BPAD + 2 * half;

  v8f acc = {};
  for (int k = 0; k < KDIM; k += 4) {
    __builtin_prefetch(aRow + k + 32, 0, 0);          // global_prefetch_b8
    v2f afrag = *(const v2f*)(aRow + k);              // A[m][k+2h .. +1]
    v2f bfrag = *(const v2f*)(bRow + k);              // Xcol[k+2h .. +1][n]
    acc = __builtin_amdgcn_wmma_f32_16x16x4_f32(
        /*neg_a=*/false, afrag, /*neg_b=*/false, bfrag,
        /*c_mod=*/(short)0, acc, /*reuse_a=*/false, /*reuse_b=*/false);
  }

  float* ob = out + b * COUT * HW;
  const int p = nt * 16 + l;
#pragma unroll
  for (int r = 0; r < 8; ++r) {
    const int cout = mt * 16 + r + 8 * half;          // ISA C/D layout: M = r + 8*half
    ob[cout * HW + p] = acc[r];
  }
}

extern "C" void kernel_launch(void* const* d_in, const int* in_sizes, int n_in,
                              void* d_out, int out_size, void* d_ws, size_t ws_size,
                              hipStream_t stream) {
  (void)in_sizes; (void)n_in; (void)out_size; (void)ws_size;
  const float* x   = (const float*)d_in[0];
  const float* wgt = (const float*)d_in[1];   // flat [COUT,CIN,3,3]
  const float* fqw = (const float*)d_in[2];
  const float* fqb = (const float*)d_in[3];
  const float* fkw = (const float*)d_in[4];
  // d_in[5] = fc_k_b: constant in n -> cancels inside the softmax, unused.
  const float* fvw = (const float*)d_in[6];
  const float* fvb = (const float*)d_in[7];
  const float* fow = (const float*)d_in[8];
  const float* fob = (const float*)d_in[9];
  const float* gam = (const float*)d_in[10];
  float* ws  = (float*)d_ws;
  float* out = (float*)d_out;

  k_sproj   <<<B_,                 256, 0, stream>>>(x, fqw, fqb, fkw, ws);
  k_wminmax <<<1,                  256, 0, stream>>>(wgt, ws);
  k_denom   <<<B_ * HW,            256, 0, stream>>>(wgt, ws);
  k_maskw   <<<dim3(WN / 256, B_), 256, 0, stream>>>(wgt, fvw, fvb, fow, fob, gam, ws);
  // 64 blocks = 16 pixel-rows x 4 batches; 8 waves/block = 8 cout tiles
  k_conv_wmma<<<dim3(16, B_),      256, LDSB, stream>>>(x, ws, out);
}